// CPCLoss_sameSeq_26723286516345
// MI455X (gfx1250) — compile-verified
//
#include <hip/hip_runtime.h>
#include <hip/hip_bf16.h>
#include <math.h>
#include <stdint.h>

typedef __attribute__((ext_vector_type(2))) float v2f;
typedef __attribute__((ext_vector_type(8))) float v8f;

#define B_     32
#define T_     1024
#define Z_     64
#define C_     256
#define K_     6
#define L_     1018          // T - K
#define LT_    128           // l-positions per workgroup
#define NCHUNK 8             // ceil(L / LT)
#define ZPAD   68            // padded row stride (floats) to rotate LDS banks

// 64-length dot of a (bank-rotated) LDS z-row against a register-resident Wc row.
__device__ __forceinline__ float dot64(const float* __restrict__ zr,
                                       const float* __restrict__ wc) {
  float f = 0.f;
#pragma unroll
  for (int q = 0; q < 16; ++q) {
    const float4 v = *(const float4*)(zr + 4 * q);
    f += v.x * wc[4 * q + 0];
    f += v.y * wc[4 * q + 1];
    f += v.z * wc[4 * q + 2];
    f += v.w * wc[4 * q + 3];
  }
  return f;
}

__launch_bounds__(256, 1)
__global__ void cpc_main(const float* __restrict__ z,
                         const float* __restrict__ c,
                         const float* __restrict__ W,
                         const float* __restrict__ bias,
                         const int*   __restrict__ seq,
                         float* __restrict__ ws) {
  __shared__ float zbuf[T_ * ZPAD];     // all of z[b]  : 1024*68*4 = 272 KB
  __shared__ float wcbuf[LT_ * ZPAD];   // Wc tile      : 128*68*4  =  34 KB
  __shared__ float red[2];

  const int tid  = threadIdx.x;
  const int kk   = blockIdx.z;          // shift index 0..5 (ref k = kk+1)
  const int bidx = blockIdx.y;          // batch
  const int l0   = blockIdx.x * LT_;    // l-chunk base

  if (tid < 2) red[tid] = 0.f;

  // ---- Stage z[b,:,:] into padded LDS via ASYNC DMA (GLOBAL_LOAD_ASYNC_TO_LDS_B128,
  //      tracked by ASYNCcnt). Fire-and-forget: the GEMM below never touches zbuf,
  //      so the copy overlaps the entire WMMA phase.
  const float* zb = z + (size_t)bidx * T_ * Z_;
  for (int i = tid; i < T_ * (Z_ / 4); i += 256) {
    const int row = i >> 4;
    const int q   = i & 15;
    const uint64_t src = (uint64_t)(uintptr_t)(zb + (size_t)row * Z_ + 4 * q);
    const uint32_t dst = (uint32_t)(uintptr_t)(&zbuf[row * ZPAD + 4 * q]); // LDS byte offset
    asm volatile("global_load_async_to_lds_b128 %0, %1, off"
                 :: "v"(dst), "v"(src)
                 : "memory");
  }

  // ---- GEMM:  Wc[l, z] = sum_c c[b, l, c] * W[kk, c, z]   (f32 WMMA 16x16x4) ----
  const int wave = tid >> 5;            // 0..7, each wave owns a 16(l) x 64(z) tile
  const int lane = tid & 31;
  const int h    = lane >> 4;           // half-wave selects K pair {0,1} / {2,3}
  const int m    = lane & 15;

  // A fragment source: c row (l0 + wave*16 + m), columns 4s + 2h + {0,1}
  const float* crow = c + ((size_t)bidx * T_ + (size_t)(l0 + wave * 16 + m)) * C_ + 2 * h;
  const float* Wk   = W + (size_t)kk * C_ * Z_;

  v8f acc0 = {}, acc1 = {}, acc2 = {}, acc3 = {};
#pragma unroll 4
  for (int s = 0; s < 64; ++s) {
    v2f a;
    a.x = crow[4 * s + 0];
    a.y = crow[4 * s + 1];
    // B fragment: rows cK = 4s + 2h + {0,1}, cols z = 16*t + m
    const float* wrow = Wk + (size_t)(4 * s + 2 * h) * Z_ + m;
    v2f b0, b1, b2, b3;
    b0.x = wrow[0];   b0.y = wrow[Z_ + 0];
    b1.x = wrow[16];  b1.y = wrow[Z_ + 16];
    b2.x = wrow[32];  b2.y = wrow[Z_ + 32];
    b3.x = wrow[48];  b3.y = wrow[Z_ + 48];
    acc0 = __builtin_amdgcn_wmma_f32_16x16x4_f32(false, a, false, b0, (short)0, acc0, false, false);
    acc1 = __builtin_amdgcn_wmma_f32_16x16x4_f32(false, a, false, b1, (short)0, acc1, false, false);
    acc2 = __builtin_amdgcn_wmma_f32_16x16x4_f32(false, a, false, b2, (short)0, acc2, false, false);
    acc3 = __builtin_amdgcn_wmma_f32_16x16x4_f32(false, a, false, b3, (short)0, acc3, false, false);
  }

  // D layout: VGPR r holds row M = r + 8*h, col N = m (per 16-wide z tile).
  // Fold in bias and the 1/sqrt(Z) = 0.125 scale while spilling to LDS.
  const float* bk = bias + kk * Z_;
#pragma unroll
  for (int r = 0; r < 8; ++r) {
    const int row = wave * 16 + 8 * h + r;
    float* wr = &wcbuf[row * ZPAD];
    wr[ 0 + m] = (acc0[r] + bk[ 0 + m]) * 0.125f;
    wr[16 + m] = (acc1[r] + bk[16 + m]) * 0.125f;
    wr[32 + m] = (acc2[r] + bk[32 + m]) * 0.125f;
    wr[48 + m] = (acc3[r] + bk[48 + m]) * 0.125f;
  }

  // Drain the async z-staging DMA, then rendezvous (also orders the wcbuf ds_stores).
  asm volatile("s_wait_asynccnt 0x0" ::: "memory");
  __syncthreads();

  // ---- Scoring: lane = l position; half-waves split the 64 negatives ----
  const int  lloc  = wave * 16 + m;           // 0..127
  const int  gl    = l0 + lloc;               // global l
  const bool valid = (gl < L_);
  const int  glc   = valid ? gl : (L_ - 1);   // clamp to stay in-bounds

  float wc[64];
#pragma unroll
  for (int q = 0; q < 16; ++q) {
    const float4 v = *(const float4*)(&wcbuf[lloc * ZPAD + 4 * q]);
    wc[4 * q + 0] = v.x; wc[4 * q + 1] = v.y;
    wc[4 * q + 2] = v.z; wc[4 * q + 3] = v.w;
  }

  // seq_index[kk, b, n, l]: l is fastest dim -> lanes (m) read coalesced.
  // Streamed exactly once grid-wide -> non-temporal so it can't evict z/c/W from L2.
  const int* sp = seq + ((((size_t)kk * B_ + bidx) * 64 + (size_t)(h * 32)) * (size_t)L_) + glc;
  float mx = -3.0e38f, sm = 0.f;   // online softmax over this half's 32 negatives
  for (int n = 0; n < 32; ++n) {
    const int sidx = __builtin_nontemporal_load(sp + (size_t)n * L_);  // in [1, L)
    int idx = sidx + glc;
    if (idx >= L_) idx -= L_;                 // (sidx + l) mod L
    const float f = dot64(&zbuf[(idx + kk + 1) * ZPAD], wc);
    if (f > mx) { sm = sm * __expf(mx - f) + 1.f; mx = f; }
    else        { sm += __expf(f - mx); }
  }

  // Merge the two half-wave partial softmaxes (lanes i <-> i^16 share the same l).
  const float mx2 = __shfl_xor(mx, 16, 32);
  const float sm2 = __shfl_xor(sm, 16, 32);
  const float Mn  = fmaxf(mx, mx2);
  const float Sn  = sm * __expf(mx - Mn) + sm2 * __expf(mx2 - Mn);

  if (h == 0 && valid) {
    const float f0 = dot64(&zbuf[(glc + kk + 1) * ZPAD], wc);  // positive sample
    const float Mt = fmaxf(Mn, f0);
    const float S  = Sn * __expf(Mn - Mt) + __expf(f0 - Mt);
    const float lossv = -(f0 - Mt - __logf(S));                // -log_softmax[0]
    const float accv  = (f0 >= Mn) ? 1.f : 0.f;                // argmax==0 (ties -> 0)
    atomicAdd(&red[0], lossv);
    atomicAdd(&red[1], accv);
  }
  __syncthreads();
  if (tid == 0) {
    atomicAdd(&ws[kk],      red[0]);
    atomicAdd(&ws[K_ + kk], red[1]);
  }
}

__global__ void cpc_finalize(const float* __restrict__ ws, float* __restrict__ out) {
  if (threadIdx.x == 0 && blockIdx.x == 0) {
    const float inv = 1.0f / (float)(B_ * L_);   // 1/32576
    float s = 0.f;
    for (int k = 0; k < K_; ++k) s += ws[k];
    out[0] = s * inv / (float)K_;                // mean over k of per-k mean NLL
    for (int k = 0; k < K_; ++k) out[1 + k] = ws[K_ + k] * inv;
  }
}

extern "C" void kernel_launch(void* const* d_in, const int* in_sizes, int n_in,
                              void* d_out, int out_size, void* d_ws, size_t ws_size,
                              hipStream_t stream) {
  const float* z    = (const float*)d_in[0];   // (B, T, Z) f32
  const float* c    = (const float*)d_in[1];   // (B, T, C) f32
  const float* W    = (const float*)d_in[2];   // (K, C, Z) f32
  const float* bias = (const float*)d_in[3];   // (K, Z)    f32
  const int*   seq  = (const int*)  d_in[4];   // (K, B, N, L) i32
  float* out = (float*)d_out;                  // [loss, acc0..acc5]
  float* ws  = (float*)d_ws;                   // 2*K partial sums

  hipMemsetAsync(ws, 0, 2 * K_ * sizeof(float), stream);

  dim3 grid(NCHUNK, B_, K_);
  cpc_main<<<grid, 256, 0, stream>>>(z, c, W, bias, seq, ws);
  cpc_finalize<<<1, 32, 0, stream>>>(ws, out);
}